// GCN_20942260535744
// MI455X (gfx1250) — compile-verified
//
#include <hip/hip_runtime.h>
#include <math.h>

typedef __attribute__((ext_vector_type(2))) float v2f;
typedef __attribute__((ext_vector_type(8))) float v8f;

#define N_FEAT  128
#define N_HID   128
#define N_CLASS 16

// ---------------------------------------------------------------------------
// GEMM1: H[n x 128] = X[n x 128] @ W[128 x 128]   (fp32 WMMA 16x16x4)
// block = 256 threads = 8 waves; wave w computes N-tile w; block strides M-tiles
// ---------------------------------------------------------------------------
__global__ __launch_bounds__(256) void gemm_xw1(const float* __restrict__ X,
                                                const float* __restrict__ W,
                                                float* __restrict__ H,
                                                int nTilesM) {
  __shared__ float sW[N_FEAT * N_HID];   // 64 KB: full W1
  __shared__ float sX[16 * N_FEAT];      // 8 KB: one M-tile of X
  const int tid = threadIdx.x;

  for (int i = tid; i < (N_FEAT * N_HID) / 4; i += 256)
    ((float4*)sW)[i] = ((const float4*)W)[i];

  const int wave = tid >> 5;
  const int lane = tid & 31;
  const int half = lane >> 4;   // 0: K=k,k+1 | rows 0..7 of C ; 1: K=k+2,k+3 | rows 8..15
  const int l16  = lane & 15;
  const int n0   = wave * 16;

  for (int mt = blockIdx.x; mt < nTilesM; mt += gridDim.x) {
    __syncthreads();
    for (int i = tid; i < (16 * N_FEAT) / 4; i += 256)
      ((float4*)sX)[i] = ((const float4*)(X + (size_t)mt * 16 * N_FEAT))[i];
    __syncthreads();

    v8f acc = {};
#pragma unroll 8
    for (int k = 0; k < N_FEAT; k += 4) {
      const int ka = k + 2 * half;
      v2f a, b;
      a.x = sX[l16 * N_FEAT + ka];
      a.y = sX[l16 * N_FEAT + ka + 1];
      b.x = sW[ka * N_HID + n0 + l16];
      b.y = sW[(ka + 1) * N_HID + n0 + l16];
      acc = __builtin_amdgcn_wmma_f32_16x16x4_f32(false, a, false, b,
                                                  (short)0, acc, false, false);
    }
    // C/D layout: VGPR r -> row (half*8 + r), col = l16
    float* out = H + ((size_t)mt * 16 + half * 8) * N_HID + n0 + l16;
#pragma unroll
    for (int r = 0; r < 8; ++r)
      out[(size_t)r * N_HID] = acc[r];
  }
}

// ---------------------------------------------------------------------------
// GEMM2: H2[n x 16] = relu(AGG + b1)[n x 128] @ W4[128 x 16]  (fused bias+ReLU)
// wave per M-tile; N=16 is a single WMMA tile
// ---------------------------------------------------------------------------
__global__ __launch_bounds__(256) void gemm_hw4(const float* __restrict__ AGG,
                                                const float* __restrict__ b1,
                                                const float* __restrict__ W4,
                                                float* __restrict__ H2,
                                                int nTilesM) {
  __shared__ float sW[N_HID * N_CLASS];  // 8 KB
  __shared__ float sB[N_HID];
  const int tid = threadIdx.x;
  for (int i = tid; i < (N_HID * N_CLASS) / 4; i += 256)
    ((float4*)sW)[i] = ((const float4*)W4)[i];
  if (tid < N_HID / 4)
    ((float4*)sB)[tid] = ((const float4*)b1)[tid];
  __syncthreads();

  const int wave = tid >> 5;
  const int lane = tid & 31;
  const int half = lane >> 4;
  const int l16  = lane & 15;

  for (int mt = blockIdx.x * 8 + wave; mt < nTilesM; mt += gridDim.x * 8) {
    const float* arow = AGG + ((size_t)mt * 16 + l16) * N_HID;
    v8f acc = {};
#pragma unroll 8
    for (int k = 0; k < N_HID; k += 4) {
      const int ka = k + 2 * half;
      v2f a, b;
      a.x = fmaxf(arow[ka]     + sB[ka],     0.0f);
      a.y = fmaxf(arow[ka + 1] + sB[ka + 1], 0.0f);
      b.x = sW[ka * N_CLASS + l16];
      b.y = sW[(ka + 1) * N_CLASS + l16];
      acc = __builtin_amdgcn_wmma_f32_16x16x4_f32(false, a, false, b,
                                                  (short)0, acc, false, false);
    }
    float* out = H2 + ((size_t)mt * 16 + half * 8) * N_CLASS + l16;
#pragma unroll
    for (int r = 0; r < 8; ++r)
      out[(size_t)r * N_CLASS] = acc[r];
  }
}

// ---------------------------------------------------------------------------
// SpMM, D=128: one wave per edge; lane l handles features 4l..4l+3
// coalesced 512B gather (global_load_b128) + 4 global_atomic_add_f32 per lane
// ---------------------------------------------------------------------------
__global__ __launch_bounds__(256) void spmm128(const float* __restrict__ val,
                                               const int* __restrict__ row,
                                               const int* __restrict__ col,
                                               const float* __restrict__ F,
                                               float* __restrict__ O,
                                               int nE) {
  const int gid    = blockIdx.x * blockDim.x + threadIdx.x;
  const int wid    = gid >> 5;
  const int lane   = gid & 31;
  const int nWaves = (gridDim.x * blockDim.x) >> 5;

  for (int e = wid; e < nE; e += nWaves) {
    const float v = val[e];
    const int   r = row[e];
    const int   c = col[e];
    // prefetch next edge's source row (speculative, L2)
    const int en = e + nWaves;
    if (en < nE)
      __builtin_prefetch(&F[(size_t)col[en] * N_FEAT + lane * 4], 0, 1);

    const float4 s = ((const float4*)(F + (size_t)c * N_FEAT))[lane];
    float* dst = O + (size_t)r * N_FEAT + lane * 4;
    atomicAdd(dst + 0, v * s.x);
    atomicAdd(dst + 1, v * s.y);
    atomicAdd(dst + 2, v * s.z);
    atomicAdd(dst + 3, v * s.w);
  }
}

// ---------------------------------------------------------------------------
// SpMM, D=16: 16 consecutive lanes per edge, one feature per lane
// ---------------------------------------------------------------------------
__global__ __launch_bounds__(256) void spmm16(const float* __restrict__ val,
                                              const int* __restrict__ row,
                                              const int* __restrict__ col,
                                              const float* __restrict__ F,
                                              float* __restrict__ O,
                                              int nE) {
  const long long total  = (long long)nE * N_CLASS;
  const long long stride = (long long)gridDim.x * blockDim.x;
  for (long long i = (long long)blockIdx.x * blockDim.x + threadIdx.x;
       i < total; i += stride) {
    const int e = (int)(i >> 4);
    const int f = (int)(i & 15);
    atomicAdd(O + (size_t)row[e] * N_CLASS + f,
              val[e] * F[(size_t)col[e] * N_CLASS + f]);
  }
}

// ---------------------------------------------------------------------------
// out = log_softmax(A + b4) over 16 classes; one thread per node row
// ---------------------------------------------------------------------------
__global__ __launch_bounds__(256) void logsoftmax16(const float* __restrict__ A,
                                                    const float* __restrict__ b4,
                                                    float* __restrict__ out,
                                                    int n) {
  const int i = blockIdx.x * blockDim.x + threadIdx.x;
  if (i >= n) return;
  float v[N_CLASS];
  float m = -3.402823466e+38f;
#pragma unroll
  for (int f = 0; f < N_CLASS; ++f) {
    v[f] = A[(size_t)i * N_CLASS + f] + b4[f];
    m = fmaxf(m, v[f]);
  }
  float s = 0.0f;
#pragma unroll
  for (int f = 0; f < N_CLASS; ++f)
    s += __expf(v[f] - m);
  const float lse = m + __logf(s);
#pragma unroll
  for (int f = 0; f < N_CLASS; ++f)
    out[(size_t)i * N_CLASS + f] = v[f] - lse;
}

// ---------------------------------------------------------------------------
extern "C" void kernel_launch(void* const* d_in, const int* in_sizes, int n_in,
                              void* d_out, int out_size, void* d_ws, size_t ws_size,
                              hipStream_t stream) {
  const float* x        = (const float*)d_in[0];
  const float* edge_val = (const float*)d_in[1];
  const float* W1       = (const float*)d_in[2];
  const float* b1       = (const float*)d_in[3];
  const float* W4       = (const float*)d_in[4];
  const float* b4       = (const float*)d_in[5];
  const int*   edge_row = (const int*)d_in[6];
  const int*   edge_col = (const int*)d_in[7];

  const int nNodes  = in_sizes[0] / N_FEAT;   // 100000
  const int nE      = in_sizes[1];            // 3200000
  const int nTilesM = nNodes / 16;            // 6250 (100000 % 16 == 0)

  // Workspace layout:
  //   h0   : nNodes*128 f32 (51.2 MB)
  //   agg1 : nNodes*128 f32 (51.2 MB)
  //   h2   : reuses first part of h0 region (nNodes*16)
  //   agg2 : reuses second half of h0 region (disjoint from h2)
  char*  ws   = (char*)d_ws;
  float* h0   = (float*)ws;
  float* agg1 = (float*)(ws + (size_t)nNodes * N_FEAT * sizeof(float));
  float* h2   = h0;
  float* agg2 = h0 + (size_t)nNodes * (N_FEAT / 2);
  float* out  = (float*)d_out;

  // 1) h0 = x @ W1
  gemm_xw1<<<1024, 256, 0, stream>>>(x, W1, h0, nTilesM);

  // 2) agg1 = A_sp @ h0
  hipMemsetAsync(agg1, 0, (size_t)nNodes * N_FEAT * sizeof(float), stream);
  spmm128<<<4096, 256, 0, stream>>>(edge_val, edge_row, edge_col, h0, agg1, nE);

  // 3) h2 = relu(agg1 + b1) @ W4
  gemm_hw4<<<782, 256, 0, stream>>>(agg1, b1, W4, h2, nTilesM);

  // 4) agg2 = A_sp @ h2
  hipMemsetAsync(agg2, 0, (size_t)nNodes * N_CLASS * sizeof(float), stream);
  spmm16<<<2048, 256, 0, stream>>>(edge_val, edge_row, edge_col, h2, agg2, nE);

  // 5) out = log_softmax(agg2 + b4)
  logsoftmax16<<<(nNodes + 255) / 256, 256, 0, stream>>>(agg2, b4, out, nNodes);
}